// SDBNExplorer_66640712565278
// MI455X (gfx1250) — compile-verified
//
#include <hip/hip_runtime.h>
#include <hip/hip_bf16.h>
#include <stdint.h>

#define DT_F 0.1f
#define PI_F 3.14159265358979323846f

// ---------------- threefry2x32-20 (JAX-compatible structure) ----------------
struct U2 { uint32_t x, y; };

__device__ __forceinline__ uint32_t rotl32(uint32_t v, int n) {
  return (v << n) | (v >> (32 - n));
}

__device__ __forceinline__ U2 threefry2x32(uint32_t k0, uint32_t k1,
                                           uint32_t c0, uint32_t c1) {
  uint32_t ks2 = k0 ^ k1 ^ 0x1BD11BDAu;
  uint32_t x0 = c0 + k0;
  uint32_t x1 = c1 + k1;
#define TF4(a, b, c, d)                                                        \
  x0 += x1; x1 = rotl32(x1, a); x1 ^= x0;                                      \
  x0 += x1; x1 = rotl32(x1, b); x1 ^= x0;                                      \
  x0 += x1; x1 = rotl32(x1, c); x1 ^= x0;                                      \
  x0 += x1; x1 = rotl32(x1, d); x1 ^= x0;
  TF4(13, 15, 26, 6)  x0 += k1;  x1 += ks2 + 1u;
  TF4(17, 29, 16, 24) x0 += ks2; x1 += k0 + 2u;
  TF4(13, 15, 26, 6)  x0 += k0;  x1 += k1 + 3u;
  TF4(17, 29, 16, 24) x0 += k1;  x1 += ks2 + 4u;
  TF4(13, 15, 26, 6)  x0 += ks2; x1 += k0 + 5u;
#undef TF4
  return {x0, x1};
}

// element idx of the flattened threefry output for a counter array of length 2n
__device__ __forceinline__ uint32_t tf_flat(uint32_t k0, uint32_t k1,
                                            uint32_t idx, uint32_t n) {
  if (idx < n) return threefry2x32(k0, k1, idx, n + idx).x;
  return threefry2x32(k0, k1, idx - n, idx).y;
}

// jax.random.split(key, n)[i]
__device__ __forceinline__ U2 jax_split(uint32_t k0, uint32_t k1,
                                        uint32_t i, uint32_t n) {
  U2 r;
  r.x = tf_flat(k0, k1, 2u * i,     n);
  r.y = tf_flat(k0, k1, 2u * i + 1, n);
  return r;
}

__device__ __forceinline__ float bits_to_u01(uint32_t b) {
  return __uint_as_float((b >> 9) | 0x3F800000u) - 1.0f;  // [0,1)
}

// jax.random.uniform(key, (N,))[b], N even
__device__ __forceinline__ float jax_u01(uint32_t k0, uint32_t k1,
                                         uint32_t b, uint32_t N) {
  return bits_to_u01(tf_flat(k0, k1, b, N >> 1));
}

// single-precision erfinv (Giles 2012 polynomial)
__device__ __forceinline__ float erfinvf_fast(float x) {
  float w = -logf((1.0f - x) * (1.0f + x));
  float p;
  if (w < 5.0f) {
    w -= 2.5f;
    p = 2.81022636e-08f;
    p = 3.43273939e-07f + p * w;  p = -3.5233877e-06f + p * w;
    p = -4.39150654e-06f + p * w; p = 0.00021858087f + p * w;
    p = -0.00125372503f + p * w;  p = -0.00417768164f + p * w;
    p = 0.246640727f + p * w;     p = 1.50140941f + p * w;
  } else {
    w = sqrtf(w) - 3.0f;
    p = -0.000200214257f;
    p = 0.000100950558f + p * w;  p = 0.00134934322f + p * w;
    p = -0.00367342844f + p * w;  p = 0.00573950773f + p * w;
    p = -0.0076224613f + p * w;   p = 0.00943887047f + p * w;
    p = 1.00167406f + p * w;      p = 2.83297682f + p * w;
  }
  return p * x;
}

// ---------------- 8-state transition-map algebra (24-bit words) -------------
__device__ __forceinline__ uint32_t compose_map(uint32_t later, uint32_t earlier) {
  uint32_t r = 0;
#pragma unroll
  for (int s = 0; s < 8; ++s) {
    uint32_t e = (earlier >> (3 * s)) & 7u;
    uint32_t v = (later >> (3 * e)) & 7u;
    r |= v << (3 * s);
  }
  return r;
}

__device__ __forceinline__ void mat3mul(float* __restrict__ C,
                                        const float* __restrict__ A,
                                        const float* __restrict__ B) {
#pragma unroll
  for (int i = 0; i < 3; ++i)
#pragma unroll
    for (int j = 0; j < 3; ++j)
      C[i * 3 + j] = A[i * 3 + 0] * B[0 * 3 + j] +
                     A[i * 3 + 1] * B[1 * 3 + j] +
                     A[i * 3 + 2] * B[2 * 3 + j];
}

// ---------------- async global->LDS (CDNA5) with fallback -------------------
#if defined(__has_builtin)
#if __has_builtin(__builtin_amdgcn_global_load_async_to_lds_b32)
#define USE_ASYNC_LDS 1
#endif
#endif

#ifdef USE_ASYNC_LDS
// Confirmed: emits global_load_async_to_lds_b32 (round-2 disasm).
typedef __attribute__((address_space(1))) int* as1_i32p;
typedef __attribute__((address_space(3))) int* as3_i32p;
#define ASYNC_CP_B32(lds_dst, g_src)                                           \
  __builtin_amdgcn_global_load_async_to_lds_b32(                               \
      (as1_i32p)(uintptr_t)(g_src),                                            \
      (as3_i32p)(uint32_t)(uintptr_t)(lds_dst), 0, 0)
__device__ __forceinline__ void wait_async_lds() {
#if __has_builtin(__builtin_amdgcn_s_wait_asynccnt)
  __builtin_amdgcn_s_wait_asynccnt(0);
#else
  asm volatile("s_wait_asynccnt 0" ::: "memory");
#endif
}
#endif

// ---------------- main simulation: one wave32 per particle ------------------
__global__ __launch_bounds__(256) void sdbn_sim(
    const float* __restrict__ x0g, const float* __restrict__ e0g,
    const float* __restrict__ e1g, const float* __restrict__ tr0g,
    const float* __restrict__ tr1g, const float* __restrict__ tr2g,
    const float* __restrict__ mug, const float* __restrict__ sgg,
    const int* __restrict__ st0g, const int* __restrict__ dtg,
    float* __restrict__ out, int B, int T) {
  __shared__ float s_tr[16];  // [0:2)=tr0, [2:6)=tr1, [6:14)=tr2
  __shared__ float s_mu[24];
  __shared__ float s_sg[24];
  __shared__ int   s_dt[24];

  const int tid = threadIdx.x;
#ifdef USE_ASYNC_LDS
  if (tid < 2)        ASYNC_CP_B32(&s_tr[tid],       tr0g + tid);
  else if (tid < 6)   ASYNC_CP_B32(&s_tr[tid],       tr1g + (tid - 2));
  else if (tid < 14)  ASYNC_CP_B32(&s_tr[tid],       tr2g + (tid - 6));
  if (tid < 24)       ASYNC_CP_B32(&s_mu[tid],       mug + tid);
  else if (tid < 48)  ASYNC_CP_B32(&s_sg[tid - 24],  sgg + (tid - 24));
  else if (tid < 72)  ASYNC_CP_B32(&s_dt[tid - 48],  dtg + (tid - 48));
  wait_async_lds();
#else
  if (tid < 2)  s_tr[tid]     = tr0g[tid];
  if (tid < 4)  s_tr[2 + tid] = tr1g[tid];
  if (tid < 8)  s_tr[6 + tid] = tr2g[tid];
  if (tid < 24) { s_mu[tid] = mug[tid]; s_sg[tid] = sgg[tid]; s_dt[tid] = dtg[tid]; }
#endif
  __syncthreads();

  const int lane = tid & 31;
  const int wave = tid >> 5;
  const int p = blockIdx.x * (blockDim.x >> 5) + wave;
  if (p >= B) return;

  // frame matrix: rows = (e0, e1, e2)
  float M[9];
  M[0] = e0g[p * 3 + 0]; M[1] = e0g[p * 3 + 1]; M[2] = e0g[p * 3 + 2];
  M[3] = e1g[p * 3 + 0]; M[4] = e1g[p * 3 + 1]; M[5] = e1g[p * 3 + 2];
  {  // e2 = normalize(cross(e0, e1))
    float cx = M[1] * M[5] - M[2] * M[4];
    float cy = M[2] * M[3] - M[0] * M[5];
    float cz = M[0] * M[4] - M[1] * M[3];
    float rn = rsqrtf(cx * cx + cy * cy + cz * cz);
    M[6] = cx * rn; M[7] = cy * rn; M[8] = cz * rn;
  }
  float xc0 = x0g[p * 3 + 0], xc1 = x0g[p * 3 + 1], xc2 = x0g[p * 3 + 2];
  int st = st0g[p * 3 + 0] * 4 + st0g[p * 3 + 1] * 2 + st0g[p * 3 + 2];

  const size_t BT = (size_t)B * (size_t)T;
  float* oX  = out + T;
  float* oS  = oX + 3 * BT;
  float* osp = oS + 3 * BT;
  float* opa = osp + BT;
  float* onp = opa + BT;
  const uint32_t Bu = (uint32_t)B, Tu = (uint32_t)T, pu = (uint32_t)p;

  for (int base = 0; base < T; base += 32) {
    const int t = base + lane;
    const int lastLane = min(31, T - base - 1);

    // per-step key: jax.random.split(key(42), T)[t]
    U2 sk = jax_split(0u, 42u, (uint32_t)t, Tu);

    // ks = split(sk, DEPTH + 6 = 9): all 18 flattened words from 9 evals
    // (flat[i] and flat[9+i] share one threefry counter pair (i, 9+i))
    uint32_t fl[18];
#pragma unroll
    for (int i = 0; i < 9; ++i) {
      U2 o = threefry2x32(sk.x, sk.y, (uint32_t)i, (uint32_t)(9 + i));
      fl[i] = o.x;
      fl[9 + i] = o.y;
    }

    float r[3];
#pragma unroll
    for (int d = 0; d < 3; ++d)
      r[d] = jax_u01(fl[2 * d], fl[2 * d + 1], pu, Bu);

    // build 8-state transition map for this step
    uint32_t mp = 0;
    {
      const int nb0[2] = {r[0] < s_tr[0] * DT_F, r[0] > s_tr[1] * DT_F};
#pragma unroll
      for (int s = 0; s < 8; ++s) {
        const int b0 = (s >> 2) & 1, b1 = (s >> 1) & 1, b2 = s & 1;
        const int v = nb0[b0];
        const int n1 = b1 ? (r[1] > s_tr[3 + 2 * v] * DT_F)
                          : (r[1] < s_tr[2 + 2 * v] * DT_F);
        const int pre = v * 2 + n1;
        const int n2 = b2 ? (r[2] > s_tr[7 + 2 * pre] * DT_F)
                          : (r[2] < s_tr[6 + 2 * pre] * DT_F);
        mp |= (uint32_t)((v << 2) | (n1 << 1) | n2) << (3 * s);
      }
    }

    // wave-inclusive scan of map composition
    uint32_t sm = mp;
#pragma unroll
    for (int d = 1; d < 32; d <<= 1) {
      uint32_t prev = __shfl_up(sm, d);
      if (lane >= d) sm = compose_map(sm, prev);
    }
    const int dec = (int)((sm >> (3 * st)) & 7u);

    // sample speed / planar / non-planar from per-state tables
    float vals[3];
#pragma unroll
    for (int pk = 0; pk < 3; ++pk) {
      const int jz = 3 + 2 * pk, ju = 4 + 2 * pk;
      float un = 2.0f * jax_u01(fl[2 * jz], fl[2 * jz + 1], pu, Bu) - 1.0f;
      un = fminf(fmaxf(un, -0.9999999f), 0.9999999f);
      const float z = 1.4142135624f * erfinvf_fast(un);
      const float u = jax_u01(fl[2 * ju], fl[2 * ju + 1], pu, Bu);
      const float m = s_mu[dec * 3 + pk];
      const float sg = s_sg[dec * 3 + pk];
      const int dtp = s_dt[dec * 3 + pk];
      const float g = m + sg * z;
      float v = (dtp == 0) ? g
              : (dtp == 1) ? __expf(g)
                           : (m + sg * __tanf(PI_F * (u - 0.5f)));
      v *= DT_F;
      if (pk > 0) v = fmodf(v, PI_F);
      vals[pk] = v;
    }
    const float speed = vals[0], pa = vals[1], npa = vals[2];

    // per-step frame update in body axes: Q = Ry(-npa) * Rz(-pa)
    float ca, sa, cb, sb;
    __sincosf(pa, &sa, &ca);
    __sincosf(npa, &sb, &cb);
    float P[9] = {cb * ca, cb * sa, -sb,
                  -sa,     ca,      0.0f,
                  sb * ca, sb * sa, cb};

    // wave-inclusive scan of 3x3 matrix products (later * earlier)
#pragma unroll
    for (int d = 1; d < 32; d <<= 1) {
      float o[9];
#pragma unroll
      for (int i = 0; i < 9; ++i) o[i] = __shfl_up(P[i], d);
      if (lane >= d) {
        float n[9];
        mat3mul(n, P, o);
#pragma unroll
        for (int i = 0; i < 9; ++i) P[i] = n[i];
      }
    }

    // only row 0 of (P * M) is needed per step: the heading e0_t
    const float ex = P[0] * M[0] + P[1] * M[3] + P[2] * M[6];
    const float ey = P[0] * M[1] + P[1] * M[4] + P[2] * M[7];
    const float ez = P[0] * M[2] + P[1] * M[5] + P[2] * M[8];
    float dx = speed * ex, dy = speed * ey, dz = speed * ez;
#pragma unroll
    for (int d = 1; d < 32; d <<= 1) {
      const float ox = __shfl_up(dx, d), oy = __shfl_up(dy, d),
                  oz = __shfl_up(dz, d);
      if (lane >= d) { dx += ox; dy += oy; dz += oz; }
    }
    const float px = xc0 + dx, py = xc1 + dy, pz = xc2 + dz;

    if (t < T) {
      const size_t o3 = ((size_t)p * T + t) * 3;
      __builtin_nontemporal_store(px, oX + o3 + 0);
      __builtin_nontemporal_store(py, oX + o3 + 1);
      __builtin_nontemporal_store(pz, oX + o3 + 2);
      __builtin_nontemporal_store((float)((dec >> 2) & 1), oS + o3 + 0);
      __builtin_nontemporal_store((float)((dec >> 1) & 1), oS + o3 + 1);
      __builtin_nontemporal_store((float)(dec & 1),        oS + o3 + 2);
      const size_t o1 = (size_t)p * T + t;
      __builtin_nontemporal_store(speed, osp + o1);
      __builtin_nontemporal_store(pa,    opa + o1);
      __builtin_nontemporal_store(npa,   onp + o1);
    }

    // carries from last valid lane
    st  = __shfl(dec, lastLane);
    xc0 = __shfl(px, lastLane);
    xc1 = __shfl(py, lastLane);
    xc2 = __shfl(pz, lastLane);
    float Pl[9];
#pragma unroll
    for (int i = 0; i < 9; ++i) Pl[i] = __shfl(P[i], lastLane);
    float Mn[9];
    mat3mul(Mn, Pl, M);
    // Gram-Schmidt renorm (matches reference's per-step renormalization)
    {
      float n0 = rsqrtf(Mn[0] * Mn[0] + Mn[1] * Mn[1] + Mn[2] * Mn[2]);
      Mn[0] *= n0; Mn[1] *= n0; Mn[2] *= n0;
      const float d01 = Mn[3] * Mn[0] + Mn[4] * Mn[1] + Mn[5] * Mn[2];
      Mn[3] -= d01 * Mn[0]; Mn[4] -= d01 * Mn[1]; Mn[5] -= d01 * Mn[2];
      const float n1 = rsqrtf(Mn[3] * Mn[3] + Mn[4] * Mn[4] + Mn[5] * Mn[5]);
      Mn[3] *= n1; Mn[4] *= n1; Mn[5] *= n1;
      Mn[6] = Mn[1] * Mn[5] - Mn[2] * Mn[4];
      Mn[7] = Mn[2] * Mn[3] - Mn[0] * Mn[5];
      Mn[8] = Mn[0] * Mn[4] - Mn[1] * Mn[3];
    }
#pragma unroll
    for (int i = 0; i < 9; ++i) M[i] = Mn[i];
  }
}

__global__ void write_ts(float* __restrict__ out, int T) {
  const int t = blockIdx.x * blockDim.x + threadIdx.x;
  if (t < T) out[t] = (float)t * DT_F;
}

extern "C" void kernel_launch(void* const* d_in, const int* in_sizes, int n_in,
                              void* d_out, int out_size, void* d_ws,
                              size_t ws_size, hipStream_t stream) {
  (void)d_ws; (void)ws_size; (void)n_in;
  const float* x0  = (const float*)d_in[0];
  const float* e0  = (const float*)d_in[1];
  const float* e1  = (const float*)d_in[2];
  const float* tr0 = (const float*)d_in[3];
  const float* tr1 = (const float*)d_in[4];
  const float* tr2 = (const float*)d_in[5];
  const float* mu  = (const float*)d_in[6];
  const float* sg  = (const float*)d_in[7];
  const int* st0   = (const int*)d_in[8];
  const int* dtp   = (const int*)d_in[9];

  const int B = in_sizes[0] / 3;
  // out_size = T + 9*B*T  ->  T = out_size / (1 + 9B)
  const int T = (int)((long long)out_size / (1LL + 9LL * (long long)B));
  float* out = (float*)d_out;

  write_ts<<<(T + 255) / 256, 256, 0, stream>>>(out, T);

  const int wavesPerBlock = 8;  // 256 threads = 8 wave32
  const int blocks = (B + wavesPerBlock - 1) / wavesPerBlock;
  sdbn_sim<<<blocks, 256, 0, stream>>>(x0, e0, e1, tr0, tr1, tr2, mu, sg, st0,
                                       dtp, out, B, T);
}